// MPS_8392366096695
// MI455X (gfx1250) — compile-verified
//
#include <hip/hip_runtime.h>

// MPS classifier, fused sweeps on gfx1250.
// N=128 sites, B=256 batch, D=64 bond, d=2 phys, C=10 classes.
//
// carry'[b,r] = x0[n,b]*(carry @ core[n,:,:,0])[b,r] + x1[n,b]*(carry @ core[n,:,:,1])[b,r]
// Combined B matrix is 64 x 128 with col c = 2r+i  (left sweep: contiguous core[n] layout)
//                                   col c = 2l+i  (right sweep: strided view, contract r)
// Column pair (2r, 2r+1) sits in adjacent lanes of the WMMA D tile, so the
// x0/x1 combine is done in-register via shfl_xor(.,1) - one barrier per step.
// B tiles are double-buffered in registers (prefetch next site during WMMAs).

#define NS    128
#define BATCH 256
#define DD    64
#define CL    10
#define CPAD  68    // carry LDS row stride (floats) - avoids DS bank conflicts

typedef __attribute__((ext_vector_type(2))) float v2f;
typedef __attribute__((ext_vector_type(8))) float v8f;

template<int DIR>
__device__ __forceinline__ void loadB(float2 (&buf)[16], const float* __restrict__ p) {
    if (DIR == 0) {
        #pragma unroll
        for (int kk = 0; kk < 16; ++kk) { buf[kk].x = p[kk * 512]; buf[kk].y = p[kk * 512 + 128]; }
    } else {
        #pragma unroll
        for (int kk = 0; kk < 16; ++kk) { buf[kk].x = p[kk * 8];   buf[kk].y = p[kk * 8 + 2]; }
    }
}

template<int DIR, int PAR>
__device__ __forceinline__ void step(int s,
                                     const float2 (&bb)[16],
                                     float (*__restrict__ carry)[16][CPAD],
                                     const float2 (*__restrict__ xs)[16],
                                     int rowA, int kh, int hi, int qcol, bool evenLane) {
    const float (*cr)[CPAD] = carry[PAR];
    float       (*cw)[CPAD] = carry[PAR ^ 1];

    v8f acc = {0.f, 0.f, 0.f, 0.f, 0.f, 0.f, 0.f, 0.f};
    #pragma unroll
    for (int kk = 0; kk < 16; ++kk) {
        const int kb = kk * 4 + kh;
        v2f a, b;
        a.x = cr[rowA][kb];
        a.y = cr[rowA][kb + 1];
        b.x = bb[kk].x;
        b.y = bb[kk].y;
        acc = __builtin_amdgcn_wmma_f32_16x16x4_f32(
            false, a, false, b, (short)0, acc, false, false);
    }

    // in-register column-pair combine: even lane holds i=0 column, its odd
    // neighbor (lane^1) holds i=1; x scalars broadcast per row from LDS
    const int ls = (DIR == 0) ? (1 + s) : (62 - s);
    #pragma unroll
    for (int i = 0; i < 8; ++i) {
        const float  other = __shfl_xor(acc[i], 1, 32);
        const float2 x     = xs[ls][hi * 8 + i];
        acc[i] = x.x * acc[i] + x.y * other;            // valid on even lanes
    }
    if (evenLane) {
        #pragma unroll
        for (int i = 0; i < 8; ++i) cw[hi * 8 + i][qcol] = acc[i];
    }
    __syncthreads();   // single barrier per step (ping-pong carry)
}

template<int DIR>
__device__ __forceinline__ void sweep_body(const float* __restrict__ core,
                                           float* __restrict__ dst,   // Al/Ar + b0*DD
                                           float (*__restrict__ carry)[16][CPAD],
                                           const float2 (*__restrict__ xs)[16],
                                           int t) {
    const int  lane = t & 31;
    const int  wv   = t >> 5;
    const int  hi   = lane >> 4;
    const int  col  = lane & 15;
    const int  c    = wv * 16 + col;        // global column 0..127
    const int  kh   = hi * 2;               // per-lane K sub-offset
    const int  rowA = col;
    const int  qcol = wv * 8 + (col >> 1);  // output column owned by even lanes
    const bool evenLane = (lane & 1) == 0;

    // Per-lane B base pointer (lane terms folded in once); literal +/-32KB per
    // site so every in-loop load keeps a compile-time instruction offset.
    const float* bp;
    if (DIR == 0) bp = core + (size_t)1   * 8192 + kh * 128 + c;
    else          bp = core + (size_t)126 * 8192 + (c >> 1) * 128 + (c & 1) + kh * 2;

    float2 bufA[16], bufB[16];
    loadB<DIR>(bufA, bp);
    bp += (DIR == 0) ? 8192 : -8192;

    for (int s = 0; s < 63; s += 2) {
        // even step: compute with bufA, prefetch bufB
        if (s + 1 < 63) { loadB<DIR>(bufB, bp); bp += (DIR == 0) ? 8192 : -8192; }
        step<DIR, 0>(s, bufA, carry, xs, rowA, kh, hi, qcol, evenLane);
        if (s + 1 >= 63) break;
        // odd step: compute with bufB, prefetch bufA
        if (s + 2 < 63) { loadB<DIR>(bufA, bp); bp += (DIR == 0) ? 8192 : -8192; }
        step<DIR, 1>(s + 1, bufB, carry, xs, rowA, kh, hi, qcol, evenLane);
    }

    // ---- write final carry (buffer 1 after 63 steps) ----
    const int bl = t >> 4;
    const int r0 = (t & 15) * 4;
    #pragma unroll
    for (int k = 0; k < 4; ++k) dst[bl * DD + r0 + k] = carry[1][bl][r0 + k];
}

__global__ __launch_bounds__(256) void mps_sweep(const float* __restrict__ inputs,
                                                 const float* __restrict__ core,
                                                 float* __restrict__ Al,
                                                 float* __restrict__ Ar) {
    __shared__ float  carry[2][16][CPAD];   // ping-pong carry, 16 x 64 (padded)
    __shared__ float2 xs[64][16];           // staged input scalars: 64 sites x 16 rows

    const int chunk = blockIdx.x & 15;      // batch chunk (16 rows)
    const int dir   = blockIdx.x >> 4;      // 0 = left sweep, 1 = right sweep
    const int b0    = chunk * 16;
    const int t     = threadIdx.x;

    const float2* inp2 = (const float2*)inputs;   // inputs[n][b] = (x0,x1) pair

    // ---- stage all x scalars this block will ever need into LDS (8 KB) ----
    {
        const int siteBase = (dir == 0) ? 0 : 64;
        #pragma unroll
        for (int e = t; e < 1024; e += 256) {
            const int site = e >> 4;
            const int bl   = e & 15;
            xs[site][bl] = inp2[(siteBase + site) * BATCH + b0 + bl];
        }
    }
    __syncthreads();

    // ---- initialize carry[0] with boundary row/column of site 0 / 127 ----
    {
        const int bl = t >> 4;
        const int r0 = (t & 15) * 4;
        const float2 x = xs[(dir == 0) ? 0 : 63][bl];
        const float* cn = core + (size_t)((dir == 0) ? 0 : (NS - 1)) * (DD * DD * 2);
        #pragma unroll
        for (int k = 0; k < 4; ++k) {
            const int r = r0 + k;
            float v;
            if (dir == 0)  // A[0][b,0,r]   = x . core[0,0,r,:]
                v = x.x * cn[r * 2 + 0]   + x.y * cn[r * 2 + 1];
            else           // A[127][b,l,0] = x . core[127,l,0,:]
                v = x.x * cn[r * 128 + 0] + x.y * cn[r * 128 + 1];
            carry[0][bl][r] = v;
        }
    }
    __syncthreads();

    if (dir == 0) sweep_body<0>(core, Al + b0 * DD, carry, xs, t);
    else          sweep_body<1>(core, Ar + b0 * DD, carry, xs, t);
}

// out[b,o] = sum_{l,r} Al[b,l] * Aout[o,l,r] * Ar[b,r]
__global__ __launch_bounds__(64) void mps_out(const float* __restrict__ Al,
                                              const float* __restrict__ Ar,
                                              const float* __restrict__ Aout,
                                              float* __restrict__ out) {
    __shared__ float arow[DD];
    __shared__ float part[CL][DD];
    const int b = blockIdx.x;
    const int l = threadIdx.x;

    arow[l] = Ar[b * DD + l];
    __syncthreads();

    const float al = Al[b * DD + l];
    #pragma unroll
    for (int o = 0; o < CL; ++o) {
        const float* Ao = Aout + ((size_t)o * DD + l) * DD;
        float s = 0.f;
        for (int r = 0; r < DD; ++r) s += Ao[r] * arow[r];
        part[o][l] = al * s;
    }
    __syncthreads();

    if (l < CL) {
        float s = 0.f;
        for (int k = 0; k < DD; ++k) s += part[l][k];
        out[b * CL + l] = s;
    }
}

extern "C" void kernel_launch(void* const* d_in, const int* in_sizes, int n_in,
                              void* d_out, int out_size, void* d_ws, size_t ws_size,
                              hipStream_t stream) {
    const float* inputs = (const float*)d_in[0];   // [128, 256, 2]
    const float* core   = (const float*)d_in[1];   // [128, 64, 64, 2]
    const float* Aout   = (const float*)d_in[2];   // [10, 64, 64]

    float* Al = (float*)d_ws;                      // [256, 64]
    float* Ar = Al + BATCH * DD;                   // [256, 64]

    // 16 batch chunks x 2 directions in one grid so both sweeps run concurrently;
    // Al/Ar -> output dependency enforced by stream order.
    mps_sweep<<<32, 256, 0, stream>>>(inputs, core, Al, Ar);
    mps_out<<<BATCH, 64, 0, stream>>>(Al, Ar, Aout, (float*)d_out);
}